// GAT1Block_4209067950798
// MI455X (gfx1250) — compile-verified
//
#include <hip/hip_runtime.h>
#include <hip/hip_bf16.h>
#include <stdint.h>

typedef __attribute__((ext_vector_type(2))) float v2f;
typedef __attribute__((ext_vector_type(8))) float v8f;

#define TILE_M 32
#define TILE_N 64
#define TILE_K 32

#define AS1 __attribute__((address_space(1)))
#define AS3 __attribute__((address_space(3)))

#if defined(__has_builtin)
#if __has_builtin(__builtin_amdgcn_global_load_async_to_lds_b32)
#define HAVE_ASYNC_LDS 1
#endif
#if __has_builtin(__builtin_amdgcn_s_wait_asynccnt)
#define HAVE_WAIT_ASYNC 1
#endif
#endif

__device__ __forceinline__ void wait_async_all() {
#if defined(HAVE_WAIT_ASYNC)
  __builtin_amdgcn_s_wait_asynccnt(0);
#else
  asm volatile("s_wait_asynccnt 0x0" ::: "memory");
#endif
}

#if defined(HAVE_ASYNC_LDS)
// b32 async copy: global -> LDS, no VGPR round trip (tracked by ASYNCcnt)
__device__ __forceinline__ void async_copy_b32(const float* gsrc, float* ldst) {
  __builtin_amdgcn_global_load_async_to_lds_b32(
      (AS1 int*)(AS1 float*)gsrc,
      (AS3 int*)(AS3 float*)ldst, 0, 0);
}
#endif

__device__ __forceinline__ float leakyf(float v, float s) { return v > 0.f ? v : v * s; }

// Order-preserving float -> uint encoding (ascending uint == ascending float)
__device__ __forceinline__ unsigned fenc(float f) {
  unsigned u = __float_as_uint(f);
  return (u & 0x80000000u) ? ~u : (u | 0x80000000u);
}
__device__ __forceinline__ float fdec(unsigned u) {
  unsigned b = (u & 0x80000000u) ? (u ^ 0x80000000u) : ~u;
  return __uint_as_float(b);
}

__global__ void fill_zero_kernel(uint32_t* __restrict__ p, long long n) {
  long long i = (long long)blockIdx.x * blockDim.x + threadIdx.x;
  if (i < n) p[i] = 0u;
}

// ---------------------------------------------------------------------------
// FP32 WMMA GEMM: C[M,N] = A[M,K] @ B[K,N] + bias[N]
// 256 threads = 8 waves; each wave owns a 16x16 tile of a 32x64 block tile.
// Full interior tiles are staged with GLOBAL_LOAD_ASYNC_TO_LDS (ASYNCcnt),
// ragged edge tiles use the zero-padding VGPR path. Inner loop is
// V_WMMA_F32_16X16X4_F32.
// ---------------------------------------------------------------------------
__global__ __launch_bounds__(256)
void wmma_gemm_bias_kernel(const float* __restrict__ A, int lda,
                           const float* __restrict__ B, int ldb,
                           const float* __restrict__ bias,
                           float* __restrict__ C, int ldc,
                           int M, int N, int K)
{
  __shared__ float As[TILE_M][TILE_K + 1];
  __shared__ float Bs[TILE_K][TILE_N + 1];

  const int bm = blockIdx.y * TILE_M;
  const int bn = blockIdx.x * TILE_N;

  const int tid  = threadIdx.x;        // 0..255
  const int wave = tid >> 5;           // 0..7
  const int lane = tid & 31;
  const int wm   = (wave >> 2) * 16;   // 0 or 16
  const int wn   = (wave & 3) * 16;    // 0,16,32,48
  const int half = lane >> 4;          // 0 or 1
  const int r    = lane & 15;

  v8f acc = {};

  for (int k0 = 0; k0 < K; k0 += TILE_K) {
    const bool fullK = (k0 + TILE_K <= K);
    const bool fullA = fullK && (bm + TILE_M <= M);
    const bool fullB = fullK && (bn + TILE_N <= N);

    // ---- Stage A tile (32x32) ----
#if defined(HAVE_ASYNC_LDS)
    if (fullA) {
      for (int i = tid; i < TILE_M * TILE_K; i += 256) {
        int row = i >> 5, col = i & 31;
        async_copy_b32(A + (size_t)(bm + row) * lda + (k0 + col), &As[row][col]);
      }
    } else
#endif
    {
      for (int i = tid; i < TILE_M * TILE_K; i += 256) {
        int row = i >> 5, col = i & 31;
        int gr = bm + row, gc = k0 + col;
        As[row][col] = (gr < M && gc < K) ? A[(size_t)gr * lda + gc] : 0.f;
      }
    }

    // ---- Stage B tile (32x64) ----
#if defined(HAVE_ASYNC_LDS)
    if (fullB) {
      for (int i = tid; i < TILE_K * TILE_N; i += 256) {
        int row = i >> 6, col = i & 63;
        async_copy_b32(B + (size_t)(k0 + row) * ldb + (bn + col), &Bs[row][col]);
      }
    } else
#endif
    {
      for (int i = tid; i < TILE_K * TILE_N; i += 256) {
        int row = i >> 6, col = i & 63;
        int gr = k0 + row, gc = bn + col;
        Bs[row][col] = (gr < K && gc < N) ? B[(size_t)gr * ldb + gc] : 0.f;
      }
    }

    // Prefetch hint for the next K chunk of B (global_prefetch_b8)
    if (k0 + TILE_K < K) {
      int pr = k0 + TILE_K + (tid >> 3);
      int pc = bn + ((tid & 7) << 3);
      if (pr < K && pc < N)
        __builtin_prefetch(&B[(size_t)pr * ldb + pc], 0, 1);
    }

#if defined(HAVE_ASYNC_LDS)
    if (fullA || fullB) wait_async_all();
#endif
    __syncthreads();

#pragma unroll
    for (int kk = 0; kk < TILE_K; kk += 4) {
      v2f a, b;
      // A 16x4 fragment: lanes 0-15 -> K=0/1, lanes 16-31 -> K=2/3
      a.x = As[wm + r][kk + 2 * half];
      a.y = As[wm + r][kk + 2 * half + 1];
      // B 4x16 fragment: same K split across lane halves
      b.x = Bs[kk + 2 * half][wn + r];
      b.y = Bs[kk + 2 * half + 1][wn + r];
      acc = __builtin_amdgcn_wmma_f32_16x16x4_f32(
          false, a, false, b, (short)0, acc, false, false);
    }
    __syncthreads();
  }

  // C/D layout: VGPR v -> row (v + 8*half), col r
#pragma unroll
  for (int v = 0; v < 8; ++v) {
    int row = bm + wm + v + 8 * half;
    int col = bn + wn + r;
    if (row < M && col < N)
      C[(size_t)row * ldc + col] = acc[v] + bias[col];
  }
}

// ---------------------------------------------------------------------------
// GATv2 edge phases. One thread per (edge, head).
// Edges: e < Ereal -> (src[e], dst[e]); else self-loop (e - Ereal).
// ---------------------------------------------------------------------------
__global__ void edge_score_kernel(const float* __restrict__ xl, const float* __restrict__ xr,
                                  int ld, const float* __restrict__ att,
                                  const int* __restrict__ esrc, const int* __restrict__ edst,
                                  int Ereal, int Etot, int H, int C,
                                  float* __restrict__ scores, unsigned* __restrict__ maxk)
{
  int t = blockIdx.x * blockDim.x + threadIdx.x;
  if (t >= Etot * H) return;
  int e = t / H, h = t - e * H;
  int s, d;
  if (e < Ereal) { s = esrc[e]; d = edst[e]; } else { s = e - Ereal; d = s; }
  const float* pl = xl + (size_t)s * ld + h * C;
  const float* pr = xr + (size_t)d * ld + h * C;
  const float* pa = att + h * C;
  float acc = 0.f;
  for (int c = 0; c < C; ++c) {
    float v = pl[c] + pr[c];
    acc += leakyf(v, 0.2f) * pa[c];
  }
  scores[t] = acc;
  atomicMax(&maxk[(size_t)d * H + h], fenc(acc));
}

__global__ void edge_exp_kernel(const int* __restrict__ edst,
                                int Ereal, int Etot, int H,
                                const unsigned* __restrict__ maxk,
                                float* __restrict__ scores,
                                float* __restrict__ den)
{
  int t = blockIdx.x * blockDim.x + threadIdx.x;
  if (t >= Etot * H) return;
  int e = t / H, h = t - e * H;
  int d = (e < Ereal) ? edst[e] : (e - Ereal);
  float m = fdec(maxk[(size_t)d * H + h]);
  if (!__builtin_isfinite(m)) m = 0.f;  // reference: non-finite max -> 0
  float ex = __expf(scores[t] - m);
  scores[t] = ex;
  atomicAdd(&den[(size_t)d * H + h], ex);
}

__global__ void edge_agg_kernel(const float* __restrict__ xl, int ld,
                                const int* __restrict__ esrc, const int* __restrict__ edst,
                                int Ereal, int Etot, int H, int C,
                                const float* __restrict__ scores,
                                const float* __restrict__ den,
                                float* __restrict__ agg)
{
  int t = blockIdx.x * blockDim.x + threadIdx.x;
  if (t >= Etot * H) return;
  int e = t / H, h = t - e * H;
  int s, d;
  if (e < Ereal) { s = esrc[e]; d = edst[e]; } else { s = e - Ereal; d = s; }
  float coeff = scores[t] / (den[(size_t)d * H + h] + 1e-16f);
  const float* pl = xl + (size_t)s * ld + h * C;
  float* pa = agg + (size_t)d * ld + h * C;
  for (int c = 0; c < C; ++c)
    atomicAdd(&pa[c], pl[c] * coeff);
}

__global__ void bias_act_kernel(float* __restrict__ buf, int ld,
                                const float* __restrict__ bias,
                                int Nn, int W, float slope)
{
  long long t = (long long)blockIdx.x * blockDim.x + threadIdx.x;
  if (t >= (long long)Nn * W) return;
  int i = (int)(t / W), j = (int)(t - (long long)i * W);
  float v = buf[(size_t)i * ld + j] + bias[j];
  buf[(size_t)i * ld + j] = leakyf(v, slope);
}

__global__ void pool_max_kernel(const float* __restrict__ h2,
                                const int* __restrict__ batch,
                                int Nn, int C, unsigned* __restrict__ poolk)
{
  long long t = (long long)blockIdx.x * blockDim.x + threadIdx.x;
  if (t >= (long long)Nn * C) return;
  int i = (int)(t / C), c = (int)(t - (long long)i * C);
  int g = batch[i];
  atomicMax(&poolk[(size_t)g * C + c], fenc(h2[(size_t)i * C + c]));
}

__global__ void pool_decode_kernel(const unsigned* __restrict__ poolk,
                                   float* __restrict__ poolf, int n)
{
  int t = blockIdx.x * blockDim.x + threadIdx.x;
  if (t >= n) return;
  float v = fdec(poolk[t]);
  if (!__builtin_isfinite(v)) v = -INFINITY;  // empty segment
  poolf[t] = v;
}

__global__ void final_act_kernel(const float* __restrict__ in,
                                 float* __restrict__ out, int n, float slope)
{
  int t = blockIdx.x * blockDim.x + threadIdx.x;
  if (t >= n) return;
  out[t] = leakyf(in[t], slope);
}

// ---------------------------------------------------------------------------
static inline int cdiv(long long a, long long b) { return (int)((a + b - 1) / b); }

extern "C" void kernel_launch(void* const* d_in, const int* in_sizes, int n_in,
                              void* d_out, int out_size, void* d_ws, size_t ws_size,
                              hipStream_t stream)
{
  // Inputs (setup_inputs order)
  const float* x     = (const float*)d_in[0];   // [N, 78]
  const int*   eidx  = (const int*)  d_in[1];   // [2, E]
  const int*   batch = (const int*)  d_in[2];   // [N]
  const float* w1l   = (const float*)d_in[3];   // [78, 390]
  const float* b1l   = (const float*)d_in[4];
  const float* w1r   = (const float*)d_in[5];
  const float* b1r   = (const float*)d_in[6];
  const float* att1  = (const float*)d_in[7];   // [5, 78]
  const float* bias1 = (const float*)d_in[8];
  const float* w2l   = (const float*)d_in[9];   // [390, 128]
  const float* b2l   = (const float*)d_in[10];
  const float* w2r   = (const float*)d_in[11];
  const float* b2r   = (const float*)d_in[12];
  const float* att2  = (const float*)d_in[13];  // [1, 128]
  const float* bias2 = (const float*)d_in[14];
  const float* fcw   = (const float*)d_in[15];  // [128, 128]
  const float* fcb   = (const float*)d_in[16];

  const int Nn    = in_sizes[0] / 78;     // 20000
  const int Ereal = in_sizes[1] / 2;      // 320000
  const int Etot  = Ereal + Nn;           // + self loops
  const int H1 = 5, C1 = 78, D1 = 390, P1 = 400;
  const int D2 = 128;
  const int NG = 64;

  const int* esrc = eidx;
  const int* edst = eidx + Ereal;

  // Workspace carve-out (floats / uints, 4B each)
  float*    xl1 = (float*)d_ws;                         // Nn*P1
  float*    xr1 = xl1 + (size_t)Nn * P1;                // Nn*P1
  float*    h1  = xr1 + (size_t)Nn * P1;                // Nn*P1 (agg1 then h1)
  float*    sc1 = h1  + (size_t)Nn * P1;                // Etot*H1
  unsigned* mk1 = (unsigned*)(sc1 + (size_t)Etot * H1); // Nn*H1
  float*    dn1 = (float*)(mk1 + (size_t)Nn * H1);      // Nn*H1
  float*    xl2 = dn1 + (size_t)Nn * H1;                // Nn*D2
  float*    xr2 = xl2 + (size_t)Nn * D2;                // Nn*D2
  float*    h2  = xr2 + (size_t)Nn * D2;                // Nn*D2 (agg2 then h2)
  float*    sc2 = h2  + (size_t)Nn * D2;                // Etot
  unsigned* mk2 = (unsigned*)(sc2 + (size_t)Etot);      // Nn
  float*    dn2 = (float*)(mk2 + (size_t)Nn);           // Nn
  unsigned* pk  = (unsigned*)(dn2 + (size_t)Nn);        // NG*D2
  float*    pf  = (float*)(pk + (size_t)NG * D2);       // NG*D2
  float*    fct = pf + (size_t)NG * D2;                 // NG*D2

  const int TB = 256;
  dim3 blk(TB);

  // ---------- Layer 1 dense transforms (WMMA) ----------
  {
    dim3 grid(cdiv(D1, TILE_N), cdiv(Nn, TILE_M));
    wmma_gemm_bias_kernel<<<grid, blk, 0, stream>>>(x, 78, w1l, D1, b1l, xl1, P1, Nn, D1, 78);
    wmma_gemm_bias_kernel<<<grid, blk, 0, stream>>>(x, 78, w1r, D1, b1r, xr1, P1, Nn, D1, 78);
  }

  // Zero agg1 (h1 buffer), mk1, dn1
  fill_zero_kernel<<<cdiv((long long)Nn * P1, TB), blk, 0, stream>>>((uint32_t*)h1, (long long)Nn * P1);
  fill_zero_kernel<<<cdiv((long long)Nn * H1 * 2, TB), blk, 0, stream>>>((uint32_t*)mk1, (long long)Nn * H1 * 2);

  // ---------- Layer 1 edge softmax + aggregate ----------
  {
    int work = Etot * H1;
    edge_score_kernel<<<cdiv(work, TB), blk, 0, stream>>>(xl1, xr1, P1, att1, esrc, edst,
                                                          Ereal, Etot, H1, C1, sc1, mk1);
    edge_exp_kernel<<<cdiv(work, TB), blk, 0, stream>>>(edst, Ereal, Etot, H1, mk1, sc1, dn1);
    edge_agg_kernel<<<cdiv(work, TB), blk, 0, stream>>>(xl1, P1, esrc, edst,
                                                        Ereal, Etot, H1, C1, sc1, dn1, h1);
    bias_act_kernel<<<cdiv((long long)Nn * D1, TB), blk, 0, stream>>>(h1, P1, bias1, Nn, D1, 0.1f);
  }

  // ---------- Layer 2 dense transforms (WMMA) ----------
  {
    dim3 grid(cdiv(D2, TILE_N), cdiv(Nn, TILE_M));
    wmma_gemm_bias_kernel<<<grid, blk, 0, stream>>>(h1, P1, w2l, D2, b2l, xl2, D2, Nn, D2, D1);
    wmma_gemm_bias_kernel<<<grid, blk, 0, stream>>>(h1, P1, w2r, D2, b2r, xr2, D2, Nn, D2, D1);
  }

  // Zero agg2 (h2 buffer), mk2, dn2
  fill_zero_kernel<<<cdiv((long long)Nn * D2, TB), blk, 0, stream>>>((uint32_t*)h2, (long long)Nn * D2);
  fill_zero_kernel<<<cdiv((long long)Nn * 2, TB), blk, 0, stream>>>((uint32_t*)mk2, (long long)Nn * 2);

  // ---------- Layer 2 edge softmax + aggregate (H=1, C=128) ----------
  {
    int work = Etot;
    edge_score_kernel<<<cdiv(work, TB), blk, 0, stream>>>(xl2, xr2, D2, att2, esrc, edst,
                                                          Ereal, Etot, 1, D2, sc2, mk2);
    edge_exp_kernel<<<cdiv(work, TB), blk, 0, stream>>>(edst, Ereal, Etot, 1, mk2, sc2, dn2);
    edge_agg_kernel<<<cdiv(work, TB), blk, 0, stream>>>(xl2, D2, esrc, edst,
                                                        Ereal, Etot, 1, D2, sc2, dn2, h2);
    bias_act_kernel<<<cdiv((long long)Nn * D2, TB), blk, 0, stream>>>(h2, D2, bias2, Nn, D2, 0.1f);
  }

  // ---------- Global max pool + FC ----------
  fill_zero_kernel<<<cdiv((long long)NG * D2, TB), blk, 0, stream>>>((uint32_t*)pk, (long long)NG * D2);
  pool_max_kernel<<<cdiv((long long)Nn * D2, TB), blk, 0, stream>>>(h2, batch, Nn, D2, pk);
  pool_decode_kernel<<<cdiv(NG * D2, TB), blk, 0, stream>>>(pk, pf, NG * D2);
  {
    dim3 grid(cdiv(D2, TILE_N), cdiv(NG, TILE_M));
    wmma_gemm_bias_kernel<<<grid, blk, 0, stream>>>(pf, D2, fcw, D2, fcb, fct, D2, NG, D2, D2);
  }
  final_act_kernel<<<cdiv(NG * D2, TB), blk, 0, stream>>>(fct, (float*)d_out, NG * D2, 0.1f);
}